// HomogeneousGNN_56899726737796
// MI455X (gfx1250) — compile-verified
//
#include <hip/hip_runtime.h>

typedef __attribute__((ext_vector_type(2))) float v2f;
typedef __attribute__((ext_vector_type(8))) float v8f;

// ---------------------------------------------------------------- elementwise
__global__ void fill_f32(float* __restrict__ p, float v, long n) {
    long i = (long)blockIdx.x * blockDim.x + threadIdx.x;
    if (i < n) p[i] = v;
}

__global__ void deg_accum(const int* __restrict__ dst, const float* __restrict__ ew,
                          float* __restrict__ deg, int e) {
    int i = blockIdx.x * blockDim.x + threadIdx.x;
    if (i < e) atomicAdd(&deg[dst[i]], ew[i]);
}

__global__ void deg_to_dinv(float* __restrict__ d, int n) {
    int i = blockIdx.x * blockDim.x + threadIdx.x;
    if (i < n) d[i] = rsqrtf(d[i]);
}

__global__ void edge_norm(const int* __restrict__ src, const int* __restrict__ dst,
                          const float* __restrict__ ew, const float* __restrict__ dinv,
                          float* __restrict__ nrm, int e) {
    int i = blockIdx.x * blockDim.x + threadIdx.x;
    if (i < e) nrm[i] = dinv[src[i]] * ew[i] * dinv[dst[i]];
}

// ------------------------------------------------------------- WMMA fp32 GEMM
// One block handles a 16-row tile of X. Each wave (32 lanes) owns one 16-wide
// column tile of the output and iterates K in steps of 4 with
// V_WMMA_F32_16X16X4_F32.  A tile staged through LDS (stride DIN+4 -> the
// 16-lane column read pattern lands on distinct banks).
template <int DIN, int DOUT>
__global__ void gemm_wmma(const float* __restrict__ X, const float* __restrict__ W,
                          float* __restrict__ H, int nrows) {
    constexpr int LDSW = DIN + 4;
    __shared__ float xs[16 * LDSW];
    const int tid  = threadIdx.x;
    const int lane = tid & 31;
    const int wave = tid >> 5;
    const long rowBase = (long)blockIdx.x * 16;

    // cooperative 128-bit loads of the 16 x DIN activation tile
    const int nvec = 16 * (DIN / 4);
    for (int idx = tid; idx < nvec; idx += blockDim.x) {
        const int r = idx / (DIN / 4);
        const int c = idx - r * (DIN / 4);
        float4 val = {0.f, 0.f, 0.f, 0.f};
        if (rowBase + r < nrows)
            val = *reinterpret_cast<const float4*>(X + (rowBase + r) * DIN + c * 4);
        float* s = &xs[r * LDSW + c * 4];
        s[0] = val.x; s[1] = val.y; s[2] = val.z; s[3] = val.w;
    }
    __syncthreads();

    // fragment addressing per CDNA5 ISA layouts (wave32)
    const int arow = lane & 15;            // A row: M = lane % 16
    const int koff = (lane >> 4) << 1;     // K pair: lanes 0-15 -> {0,1}, 16-31 -> {2,3}
    const int col  = wave * 16 + (lane & 15);  // B/C/D column

    v8f acc = {};
    for (int k0 = 0; k0 < DIN; k0 += 4) {
        v2f a, b;
        a.x = xs[arow * LDSW + k0 + koff];
        a.y = xs[arow * LDSW + k0 + koff + 1];
        b.x = W[(long)(k0 + koff) * DOUT + col];
        b.y = W[(long)(k0 + koff + 1) * DOUT + col];
        acc = __builtin_amdgcn_wmma_f32_16x16x4_f32(
            /*neg_a=*/false, a, /*neg_b=*/false, b,
            /*c_mod=*/(short)0, acc, /*reuse_a=*/false, /*reuse_b=*/false);
    }

    // D: VGPR v holds row v (lanes 0-15) / row v+8 (lanes 16-31)
    const int mbase = (lane < 16) ? 0 : 8;
#pragma unroll
    for (int v = 0; v < 8; ++v) {
        const long m = rowBase + mbase + v;
        if (m < nrows) H[m * DOUT + col] = acc[v];
    }
}

// ----------------------------------------------------------- scatter-add agg
// 2^LOG2D consecutive lanes share one edge: contiguous gather of the h[src]
// row and same-cacheline float atomics on agg[dst].
template <int LOG2D>
__global__ void aggregate(const float* __restrict__ h, const float* __restrict__ nrm,
                          const int* __restrict__ src, const int* __restrict__ dst,
                          float* __restrict__ agg, long total) {
    long tid = (long)blockIdx.x * blockDim.x + threadIdx.x;
    if (tid >= total) return;
    const int e = (int)(tid >> LOG2D);
    const int f = (int)(tid & ((1 << LOG2D) - 1));
    const float val = h[((long)src[e] << LOG2D) + f] * nrm[e];
    atomicAdd(&agg[((long)dst[e] << LOG2D) + f], val);
}

// ----------------------------------------------- combine: self-loop+bias+relu
template <int LOG2D, bool RELU>
__global__ void combine(const float* __restrict__ agg, const float* __restrict__ h,
                        const float* __restrict__ dinv, const float* __restrict__ bias,
                        float* __restrict__ out, long total) {
    long tid = (long)blockIdx.x * blockDim.x + threadIdx.x;
    if (tid >= total) return;
    const long i = tid >> LOG2D;
    const int  f = (int)(tid & ((1 << LOG2D) - 1));
    const float di = dinv[i];
    float v = agg[tid] + h[tid] * (di * di) + bias[f];
    out[tid] = RELU ? fmaxf(v, 0.f) : v;
}

// --------------------------------------- layer-3 combine + log_softmax fused
__global__ void final_logsoftmax(const float* __restrict__ agg, const float* __restrict__ h,
                                 const float* __restrict__ dinv, const float* __restrict__ bias,
                                 float* __restrict__ out, int n) {
    int i = blockIdx.x * blockDim.x + threadIdx.x;
    if (i >= n) return;
    const float di = dinv[i];
    const float d2 = di * di;
    float v[16];
    float m = -INFINITY;
#pragma unroll
    for (int k = 0; k < 16; ++k) {
        v[k] = agg[(long)i * 16 + k] + h[(long)i * 16 + k] * d2 + bias[k];
        m = fmaxf(m, v[k]);
    }
    float s = 0.f;
#pragma unroll
    for (int k = 0; k < 16; ++k) s += expf(v[k] - m);
    const float l = logf(s);
#pragma unroll
    for (int k = 0; k < 16; ++k) out[(long)i * 16 + k] = v[k] - m - l;
}

static inline int gridFor(long n, int b) { return (int)((n + b - 1) / b); }

extern "C" void kernel_launch(void* const* d_in, const int* in_sizes, int n_in,
                              void* d_out, int out_size, void* d_ws, size_t ws_size,
                              hipStream_t stream) {
    (void)n_in; (void)out_size; (void)ws_size;

    const float* x   = (const float*)d_in[0];
    const int*   ei  = (const int*)d_in[1];     // [2, E] (JAX default int32)
    const float* ew  = (const float*)d_in[2];
    const float* W1  = (const float*)d_in[3];
    const float* b1  = (const float*)d_in[4];
    const float* W2  = (const float*)d_in[5];
    const float* b2  = (const float*)d_in[6];
    const float* W3  = (const float*)d_in[7];
    const float* b3  = (const float*)d_in[8];
    float* out = (float*)d_out;

    const int N = in_sizes[0] / 128;
    const int E = in_sizes[2];
    const int* src = ei;
    const int* dst = ei + E;

    // workspace layout (floats)
    float* ws     = (float*)d_ws;
    float* dinv   = ws;                              // N
    float* nrm    = dinv + N;                        // E
    float* hbuf   = nrm + E;                         // N*64
    float* aggbuf = hbuf + (size_t)N * 64;           // N*64
    float* actbuf = aggbuf + (size_t)N * 64;         // N*64

    const int B = 256;

    // --- degree / norm precompute (layer-invariant) ---
    fill_f32<<<gridFor(N, B), B, 0, stream>>>(dinv, 1.0f, N);              // self-loop
    deg_accum<<<gridFor(E, B), B, 0, stream>>>(dst, ew, dinv, E);
    deg_to_dinv<<<gridFor(N, B), B, 0, stream>>>(dinv, N);
    edge_norm<<<gridFor(E, B), B, 0, stream>>>(src, dst, ew, dinv, nrm, E);

    const int rowTiles = (N + 15) / 16;

    // --- layer 1: 128 -> 64, ReLU ---
    gemm_wmma<128, 64><<<rowTiles, 4 * 32, 0, stream>>>(x, W1, hbuf, N);
    fill_f32<<<gridFor((long)N * 64, B), B, 0, stream>>>(aggbuf, 0.f, (long)N * 64);
    aggregate<6><<<gridFor((long)E * 64, B), B, 0, stream>>>(hbuf, nrm, src, dst, aggbuf, (long)E * 64);
    combine<6, true><<<gridFor((long)N * 64, B), B, 0, stream>>>(aggbuf, hbuf, dinv, b1, actbuf, (long)N * 64);

    // --- layer 2: 64 -> 64, ReLU ---
    gemm_wmma<64, 64><<<rowTiles, 4 * 32, 0, stream>>>(actbuf, W2, hbuf, N);
    fill_f32<<<gridFor((long)N * 64, B), B, 0, stream>>>(aggbuf, 0.f, (long)N * 64);
    aggregate<6><<<gridFor((long)E * 64, B), B, 0, stream>>>(hbuf, nrm, src, dst, aggbuf, (long)E * 64);
    combine<6, true><<<gridFor((long)N * 64, B), B, 0, stream>>>(aggbuf, hbuf, dinv, b2, actbuf, (long)N * 64);

    // --- layer 3: 64 -> 16, then log_softmax ---
    gemm_wmma<64, 16><<<rowTiles, 1 * 32, 0, stream>>>(actbuf, W3, hbuf, N);
    fill_f32<<<gridFor((long)N * 16, B), B, 0, stream>>>(aggbuf, 0.f, (long)N * 16);
    aggregate<4><<<gridFor((long)E * 16, B), B, 0, stream>>>(hbuf, nrm, src, dst, aggbuf, (long)E * 16);
    final_logsoftmax<<<gridFor(N, B), B, 0, stream>>>(aggbuf, hbuf, dinv, b3, out, N);
}